// GraphAttentionLayer_67534065762580
// MI455X (gfx1250) — compile-verified
//
#include <hip/hip_runtime.h>

// GAT layer for MI455X (gfx1250, wave32, WMMA).
// f32 path via V_WMMA_F32_16X16X4_F32 to match the f32 reference exactly.
// Round 2: 4-way A-fragment reuse across N-tiles in both WMMA GEMMs.

typedef float v2f __attribute__((ext_vector_type(2)));
typedef float v8f __attribute__((ext_vector_type(8)));
typedef int   v2i __attribute__((ext_vector_type(2)));

#define GAT_ALPHA 0.2f
#define GAT_NEG_INF -9.0e15f

constexpr int Bb   = 8;
constexpr int Nn   = 2048;
constexpr int FIN  = 512;
constexpr int FOUT = 256;

__device__ __forceinline__ float lrelu(float x) {
    return x > 0.f ? x : GAT_ALPHA * x;
}

__device__ __forceinline__ v8f wmma_f32(v2f a, v2f b, v8f c) {
    return __builtin_amdgcn_wmma_f32_16x16x4_f32(false, a, false, b,
                                                 (short)0, c, false, false);
}

// ---------------------------------------------------------------------------
// Kernel 1: h = x @ W    (B x N x FIN) @ (FIN x FOUT) -> (B x N x FOUT)
// 4 waves / block; each wave computes a 16x64 strip (4 C tiles) sharing one
// A fragment per k-step. Block covers 16 rows x all 256 cols.
// ---------------------------------------------------------------------------
__global__ __launch_bounds__(128) void k_gemm_h(const float* __restrict__ x,
                                                const float* __restrict__ W,
                                                float* __restrict__ h) {
    const int itile = blockIdx.x;          // 0..127 (16-row tile)
    const int b     = blockIdx.y;
    const int wave  = threadIdx.x >> 5;    // 0..3
    const int lane  = threadIdx.x & 31;
    const int m     = lane & 15;           // A row / B,C col within tile
    const int koff  = (lane >> 4) << 1;    // K sub-offset (0 or 2)
    const int i0    = itile * 16;
    const int n0    = wave * 64;           // wave owns 64 output cols

    const float* xrow = x + ((long)b * Nn + (i0 + m)) * FIN;
    const float* Wc   = W + n0 + m;

    v8f c0 = {}, c1 = {}, c2 = {}, c3 = {};
    #pragma unroll 4
    for (int k0 = 0; k0 < FIN; k0 += 4) {
        const int k = k0 + koff;
        // A frag: x[b, i0+m, k..k+1]  (contiguous, 8B aligned) — shared by 4 WMMAs
        const v2f a = *(const v2f*)(xrow + k);
        const float* w0 = Wc + (long)k * FOUT;
        const float* w1 = w0 + FOUT;
        const v2f bf0 = {w0[0],  w1[0]};
        const v2f bf1 = {w0[16], w1[16]};
        const v2f bf2 = {w0[32], w1[32]};
        const v2f bf3 = {w0[48], w1[48]};
        c0 = wmma_f32(a, bf0, c0);
        c1 = wmma_f32(a, bf1, c1);
        c2 = wmma_f32(a, bf2, c2);
        c3 = wmma_f32(a, bf3, c3);
    }

    // C layout: VGPR r, lanes 0-15 -> M=r, lanes 16-31 -> M=r+8; N = lane&15
    float* hout = h + ((long)b * Nn + i0) * FOUT + n0 + m;
    const int rbase = (lane >> 4) * 8;
    #pragma unroll
    for (int r = 0; r < 8; ++r) {
        float* orow = hout + (long)(rbase + r) * FOUT;
        orow[0]  = c0[r];
        orow[16] = c1[r];
        orow[32] = c2[r];
        orow[48] = c3[r];
    }
}

// ---------------------------------------------------------------------------
// Kernel 2: s = h @ a1, t = h @ a2   — one wave per (b,i) row of h
// ---------------------------------------------------------------------------
__global__ __launch_bounds__(256) void k_st(const float* __restrict__ h,
                                            const float* __restrict__ a,
                                            float* __restrict__ s,
                                            float* __restrict__ t) {
    const int row  = blockIdx.x * 8 + (threadIdx.x >> 5);   // b*Nn + i
    const int lane = threadIdx.x & 31;
    const float* hr = h + (long)row * FOUT;

    float accs = 0.f, acct = 0.f;
    #pragma unroll
    for (int o = lane; o < FOUT; o += 32) {
        const float hv = hr[o];
        accs += hv * a[o];
        acct += hv * a[FOUT + o];
    }
    #pragma unroll
    for (int off = 16; off > 0; off >>= 1) {
        accs += __shfl_xor(accs, off, 32);
        acct += __shfl_xor(acct, off, 32);
    }
    if (lane == 0) { s[row] = accs; t[row] = acct; }
}

// ---------------------------------------------------------------------------
// Kernel 3: per-row online softmax stats over j:
//   e_ij = adj>0 ? lrelu(s_i + t_j) : NEG_INF ; rowmax, rowinv = 1/sum(exp(e-max))
// One wave per row; exact NEG_INF semantics so fully-masked rows match reference.
// ---------------------------------------------------------------------------
__global__ __launch_bounds__(256) void k_softmax_stats(const int* __restrict__ adj,
                                                       const float* __restrict__ s,
                                                       const float* __restrict__ t,
                                                       float* __restrict__ rowmax,
                                                       float* __restrict__ rowinv) {
    const int row  = blockIdx.x * 8 + (threadIdx.x >> 5);   // b*Nn + i
    const int lane = threadIdx.x & 31;
    const int b    = row >> 11;                             // / Nn
    const int* arow = adj + (long)row * Nn;
    const float* tb = t + (long)b * Nn;
    const float sv  = s[row];

    float m = -__builtin_inff();
    float l = 0.f;
    for (int j = lane; j < Nn; j += 32) {
        const float e  = (arow[j] > 0) ? lrelu(sv + tb[j]) : GAT_NEG_INF;
        const float mn = fmaxf(m, e);
        l = l * __expf(m - mn) + __expf(e - mn);
        m = mn;
    }
    #pragma unroll
    for (int off = 16; off > 0; off >>= 1) {
        const float mo = __shfl_xor(m, off, 32);
        const float lo = __shfl_xor(l, off, 32);
        const float mn = fmaxf(m, mo);
        l = l * __expf(m - mn) + lo * __expf(mo - mn);
        m = mn;
    }
    if (lane == 0) { rowmax[row] = m; rowinv[row] = 1.f / l; }
}

// ---------------------------------------------------------------------------
// Kernel 4: out = elu( softmax(e) @ h )  — flash-style: P fragments built on
// the fly (exp(e - rowmax)), normalized by rowinv in the epilogue.
// 4 waves / block; each wave owns a 16x64 strip (4 C tiles) so each P
// fragment (adj load + 2 exps) feeds 4 WMMAs.
// ---------------------------------------------------------------------------
__global__ __launch_bounds__(128) void k_attn_gemm(const int* __restrict__ adj,
                                                   const float* __restrict__ h,
                                                   const float* __restrict__ s,
                                                   const float* __restrict__ t,
                                                   const float* __restrict__ rowmax,
                                                   const float* __restrict__ rowinv,
                                                   float* __restrict__ out) {
    __shared__ float inv_s[16];
    const int itile = blockIdx.x;          // 0..127
    const int b     = blockIdx.y;
    const int wave  = threadIdx.x >> 5;    // 0..3
    const int lane  = threadIdx.x & 31;
    const int m     = lane & 15;
    const int koff  = (lane >> 4) << 1;
    const int i0    = itile * 16;
    const long rowbase = (long)b * Nn + i0;

    if (threadIdx.x < 16) inv_s[threadIdx.x] = rowinv[rowbase + threadIdx.x];
    __syncthreads();

    const float sv = s[rowbase + m];
    const float mm = rowmax[rowbase + m];
    const int*   arow = adj + ((long)b * Nn + (i0 + m)) * Nn;
    const float* tb   = t + (long)b * Nn;
    const float* hb   = h + (long)b * Nn * FOUT;
    const int f0 = wave * 64;              // wave owns 64 output cols

    v8f c0 = {}, c1 = {}, c2 = {}, c3 = {};
    #pragma unroll 2
    for (int j0 = 0; j0 < Nn; j0 += 4) {
        const int j = j0 + koff;
        // P fragment (A): exp(e - rowmax) for (row=i0+m, cols j, j+1)
        const v2i ad = *(const v2i*)(arow + j);            // 8B-aligned
        const float t0 = tb[j], t1 = tb[j + 1];
        const float e0 = (ad.x > 0) ? lrelu(sv + t0) : GAT_NEG_INF;
        const float e1 = (ad.y > 0) ? lrelu(sv + t1) : GAT_NEG_INF;
        v2f a;
        a.x = __expf(e0 - mm);
        a.y = __expf(e1 - mm);
        // H fragments (B): rows j, j+1 of h, four 16-col tiles
        const float* hrow0 = hb + (long)j * FOUT + f0 + m;
        const float* hrow1 = hrow0 + FOUT;
        const v2f bf0 = {hrow0[0],  hrow1[0]};
        const v2f bf1 = {hrow0[16], hrow1[16]};
        const v2f bf2 = {hrow0[32], hrow1[32]};
        const v2f bf3 = {hrow0[48], hrow1[48]};
        c0 = wmma_f32(a, bf0, c0);
        c1 = wmma_f32(a, bf1, c1);
        c2 = wmma_f32(a, bf2, c2);
        c3 = wmma_f32(a, bf3, c3);
    }

    // Epilogue: normalize by 1/sum per output row, then ELU (alpha=1).
    const int rbase = (lane >> 4) * 8;
    #pragma unroll
    for (int r = 0; r < 8; ++r) {
        const int rr   = rbase + r;
        const float in = inv_s[rr];
        float v0 = c0[r] * in;
        float v1 = c1[r] * in;
        float v2 = c2[r] * in;
        float v3 = c3[r] * in;
        v0 = v0 > 0.f ? v0 : (__expf(v0) - 1.f);
        v1 = v1 > 0.f ? v1 : (__expf(v1) - 1.f);
        v2 = v2 > 0.f ? v2 : (__expf(v2) - 1.f);
        v3 = v3 > 0.f ? v3 : (__expf(v3) - 1.f);
        float* orow = out + (rowbase + rr) * FOUT + f0 + m;
        orow[0]  = v0;
        orow[16] = v1;
        orow[32] = v2;
        orow[48] = v3;
    }
}

// ---------------------------------------------------------------------------
extern "C" void kernel_launch(void* const* d_in, const int* in_sizes, int n_in,
                              void* d_out, int out_size, void* d_ws, size_t ws_size,
                              hipStream_t stream) {
    const float* x   = (const float*)d_in[0];   // (8,2048,512)
    const int*   adj = (const int*)  d_in[1];   // (8,2048,2048)
    const float* W   = (const float*)d_in[2];   // (512,256)
    const float* a   = (const float*)d_in[3];   // (512,1)
    float* out = (float*)d_out;                 // (8,2048,256)

    char* ws = (char*)d_ws;
    float* h = (float*)ws;                                   // B*N*FOUT f32
    const size_t hbytes = (size_t)Bb * Nn * FOUT * sizeof(float);
    float* s    = (float*)(ws + hbytes);                     // B*N
    float* t    = s + (size_t)Bb * Nn;                       // B*N
    float* rmax = t + (size_t)Bb * Nn;                       // B*N
    float* rinv = rmax + (size_t)Bb * Nn;                    // B*N

    k_gemm_h       <<<dim3(Nn / 16, Bb), 128, 0, stream>>>(x, W, h);
    k_st           <<<dim3(Bb * Nn / 8), 256, 0, stream>>>(h, a, s, t);
    k_softmax_stats<<<dim3(Bb * Nn / 8), 256, 0, stream>>>(adj, s, t, rmax, rinv);
    k_attn_gemm    <<<dim3(Nn / 16, Bb), 128, 0, stream>>>(adj, h, s, t, rmax, rinv, out);
}